// EMDCriterion_3264175145042
// MI455X (gfx1250) — compile-verified
//
#include <hip/hip_runtime.h>

// ---------------------------------------------------------------------------
// Sinkhorn EMD criterion for MI455X (gfx1250, wave32).
// Cost-matrix tiles are recomputed on the fly with V_WMMA_F32_16X16X4_F32
// (K=3 padded to 4), fused with an online log-sum-exp, so the 256MB cost
// matrix never exists in memory. Points are pre-packed as float4
// {x,y,z,||p||^2} so each column tile is one global_load_b128; column mask
// bias is folded into the dual at write time so the inner loop does exactly
// two loads per tile per lane. j-loop unrolled 4x for load/WMMA overlap.
// ---------------------------------------------------------------------------

typedef __attribute__((ext_vector_type(2))) float v2f;
typedef __attribute__((ext_vector_type(8))) float v8f;

#define B_   16
#define N_   2048
#define L_   7
#define TIL  16
#define NTILES (N_ / TIL)          // 128 tiles per row/col dimension
#define INV_EPS 50.0f              // 1 / SINKHORN_EPS (0.02)
#define NEGV  (-1.0e9f)
#define NITER 20

// ---------------- pack points: {p0,p1,p2, squared norm} ------------------
__global__ __launch_bounds__(256) void k_prep(
    const float* __restrict__ x, const float* __restrict__ y,
    float4* __restrict__ xpack, float4* __restrict__ ypack) {
  int idx = blockIdx.x * 256 + threadIdx.x;
  if (idx < B_ * N_) {
    const float* p = x + (size_t)idx * 3;
    xpack[idx] = make_float4(p[0], p[1], p[2], p[0]*p[0] + p[1]*p[1] + p[2]*p[2]);
    const float* q = y + (size_t)idx * 3;
    ypack[idx] = make_float4(q[0], q[1], q[2], q[0]*q[0] + q[1]*q[1] + q[2]*q[2]);
  }
}

// ---------------- valid-point counts per batch ---------------------------
__global__ void k_counts(const unsigned char* __restrict__ xm,
                         const unsigned char* __restrict__ ym,
                         float* __restrict__ na, float* __restrict__ nb) {
  int b = blockIdx.x;            // 16 blocks x 32 threads (one wave)
  int lane = threadIdx.x;
  int ca = 0, cb = 0;
  for (int i = lane; i < N_; i += 32) {
    ca += xm[b * N_ + i] ? 1 : 0;
    cb += ym[b * N_ + i] ? 1 : 0;
  }
  for (int off = 16; off; off >>= 1) {
    ca += __shfl_xor(ca, off, 32);
    cb += __shfl_xor(cb, off, 32);
  }
  if (lane == 0) { na[b] = (float)ca; nb[b] = (float)cb; }
}

// ---------------- init duals (mask-folded) + accumulator -----------------
__global__ __launch_bounds__(256) void k_zero(
    float* __restrict__ u, float* __restrict__ v,
    const unsigned char* __restrict__ ym, float* __restrict__ acc) {
  int idx = blockIdx.x * 256 + threadIdx.x;
  if (idx < B_ * N_) {
    u[idx] = 0.0f;                                 // overwritten before use
    v[idx] = ym[idx] ? 0.0f : NEGV;                // v=0 with col-mask folded in
  }
  if (idx == 0) acc[0] = 0.0f;
}

// ---------------- one Sinkhorn half-step (u OR v update) -----------------
// dual_out[i] = rowmask_i + log_marg_row[i]
//             - logsumexp_j( -C(i,j)/eps + rowmask_i + dual_eff_col[j] )
// dual_eff already contains the column mask bias. Row/col symmetric, so
// the v-step reuses this kernel with swapped arguments.
__global__ __launch_bounds__(256) void k_step(
    const float4* __restrict__ rowpack, const unsigned char* __restrict__ Mrow,
    const float* __restrict__ cntrow,
    const float4* __restrict__ colpack, const float* __restrict__ dualcol,
    float* __restrict__ dualrow_out) {
  const int lane  = threadIdx.x & 31;
  const int wid   = threadIdx.x >> 5;
  const int tile  = blockIdx.x * 8 + wid;      // 0..2047, exact cover
  const int b     = tile >> 7;                 // / NTILES
  const int i0    = (tile & (NTILES - 1)) * TIL;
  const int half  = lane >> 4;                 // 0: lanes 0-15, 1: lanes 16-31
  const int l16   = lane & 15;

  // A fragment: 16x4 f32, rows i0..i0+15. Lanes 0-15 hold K=0,1; 16-31 K=2,pad.
  const float4 ap = rowpack[(size_t)b * N_ + i0 + l16];
  v2f afrag;
  afrag.x = half ? ap.z : ap.x;
  afrag.y = half ? 0.0f : ap.y;

  // Per-lane row metadata. C/D layout: lanes 0-15 hold rows r, lanes 16-31 r+8.
  const int rbase = i0 + half * 8;
  float rn2[8], rbias[8], loga[8];
  const float lncnt = __logf(cntrow[b]);
#pragma unroll
  for (int r = 0; r < 8; ++r) {
    int row = rbase + r;
    rn2[r] = rowpack[(size_t)b * N_ + row].w;
    bool valid = Mrow[b * N_ + row] != 0;
    rbias[r] = valid ? 0.0f : NEGV;
    loga[r]  = valid ? -lncnt : NEGV;
  }

  float mx[8], sm[8];
#pragma unroll
  for (int r = 0; r < 8; ++r) { mx[r] = -3.0e38f; sm[r] = 0.0f; }

#pragma unroll 4
  for (int jt = 0; jt < NTILES; ++jt) {
    const int j = jt * TIL + l16;
    const float4 cp = colpack[(size_t)b * N_ + j];   // one b128 load
    const float  cb = dualcol[b * N_ + j];           // one b32 load (mask folded)
    v2f bfrag;                                       // 4x16 f32 B fragment
    bfrag.x = half ? cp.z : cp.x;
    bfrag.y = half ? 0.0f : cp.y;
    const float cn2 = cp.w;

    v8f acc = {0.f, 0.f, 0.f, 0.f, 0.f, 0.f, 0.f, 0.f};
    acc = __builtin_amdgcn_wmma_f32_16x16x4_f32(
        false, afrag, false, bfrag, (short)0, acc, false, false);

#pragma unroll
    for (int r = 0; r < 8; ++r) {
      float C = fmaxf(rn2[r] + cn2 - 2.0f * acc[r], 0.0f);
      float m = -C * INV_EPS + rbias[r] + cb;        // M + dual_eff
      float nm = fmaxf(mx[r], m);
      sm[r] = sm[r] * __expf(mx[r] - nm) + __expf(m - nm);
      mx[r] = nm;
    }
  }

  // Merge (max, sumexp) across the 16 lanes of each half-wave.
#pragma unroll
  for (int r = 0; r < 8; ++r) {
#pragma unroll
    for (int off = 1; off < 16; off <<= 1) {
      float omx = __shfl_xor(mx[r], off, 32);
      float osm = __shfl_xor(sm[r], off, 32);
      float nm = fmaxf(mx[r], omx);
      sm[r] = sm[r] * __expf(mx[r] - nm) + osm * __expf(omx - nm);
      mx[r] = nm;
    }
  }
  if (l16 == 0) {
#pragma unroll
    for (int r = 0; r < 8; ++r) {
      float lse = mx[r] + __logf(sm[r]);
      // store dual with the row-mask bias folded in (dual_eff for next step)
      dualrow_out[b * N_ + rbase + r] = (loga[r] - lse) + rbias[r];
    }
  }
}

// ---------------- final sum_{b,i,j} P*C  ---------------------------------
// P = exp(M + u + v); u,v already contain their mask biases (= M's biases).
__global__ __launch_bounds__(256) void k_pc(
    const float4* __restrict__ rowpack, const float4* __restrict__ colpack,
    const float* __restrict__ u, const float* __restrict__ v,
    float* __restrict__ acc_out) {
  const int lane  = threadIdx.x & 31;
  const int wid   = threadIdx.x >> 5;
  const int tile  = blockIdx.x * 8 + wid;
  const int b     = tile >> 7;
  const int i0    = (tile & (NTILES - 1)) * TIL;
  const int half  = lane >> 4;
  const int l16   = lane & 15;

  const float4 ap = rowpack[(size_t)b * N_ + i0 + l16];
  v2f afrag;
  afrag.x = half ? ap.z : ap.x;
  afrag.y = half ? 0.0f : ap.y;

  const int rbase = i0 + half * 8;
  float rn2[8], rb[8];
#pragma unroll
  for (int r = 0; r < 8; ++r) {
    int row = rbase + r;
    rn2[r] = rowpack[(size_t)b * N_ + row].w;
    rb[r]  = u[b * N_ + row];                        // u_eff (mask folded)
  }

  float s = 0.0f;
#pragma unroll 4
  for (int jt = 0; jt < NTILES; ++jt) {
    const int j = jt * TIL + l16;
    const float4 cp = colpack[(size_t)b * N_ + j];
    const float  cb = v[b * N_ + j];                 // v_eff (mask folded)
    v2f bfrag;
    bfrag.x = half ? cp.z : cp.x;
    bfrag.y = half ? 0.0f : cp.y;
    const float cn2 = cp.w;

    v8f acc = {0.f, 0.f, 0.f, 0.f, 0.f, 0.f, 0.f, 0.f};
    acc = __builtin_amdgcn_wmma_f32_16x16x4_f32(
        false, afrag, false, bfrag, (short)0, acc, false, false);

#pragma unroll
    for (int r = 0; r < 8; ++r) {
      float C = fmaxf(rn2[r] + cn2 - 2.0f * acc[r], 0.0f);
      float m = -C * INV_EPS + rb[r] + cb;
      s += __expf(m) * C;
    }
  }
  for (int off = 1; off < 32; off <<= 1) s += __shfl_xor(s, off, 32);
  if (lane == 0) atomicAdd(acc_out, s);
}

// ---------------- kl / beta / output assembly ----------------------------
__global__ void k_fin(const float* __restrict__ kls, const int* __restrict__ epoch,
                      const float* __restrict__ l1acc, float* __restrict__ out) {
  if (threadIdx.x == 0 && blockIdx.x == 0) {
    const float zs[L_] = {1.f, 1.f, 2.f, 4.f, 8.f, 16.f, 32.f};
    float col[L_]; float tot = 0.0f;
    for (int l = 0; l < L_; ++l) {
      float c = 0.0f;
      for (int b = 0; b < B_; ++b) c += kls[b * L_ + l];
      col[l] = c; tot += c;
    }
    float kl = tot / (float)B_;
    float w = (float)epoch[0] / 50.0f; if (w > 1.0f) w = 1.0f;
    float beta = 0.01f * w;
    float l1 = l1acc[0] / (float)B_;
    out[0] = beta * kl + l1;     // loss
    out[1] = kl;                 // kl_loss
    out[2] = l1;                 // l1_loss
    for (int l = 0; l < L_; ++l) out[3 + l] = (col[l] / (float)B_) / (zs[l] * 16.0f);
    out[10] = beta;
  }
}

// ---------------------------------------------------------------------------
extern "C" void kernel_launch(void* const* d_in, const int* in_sizes, int n_in,
                              void* d_out, int out_size, void* d_ws, size_t ws_size,
                              hipStream_t stream) {
  const float*         x     = (const float*)d_in[0];
  const unsigned char* xm    = (const unsigned char*)d_in[1];
  const float*         y     = (const float*)d_in[2];
  const unsigned char* ym    = (const unsigned char*)d_in[3];
  const float*         kls   = (const float*)d_in[4];
  const int*           epoch = (const int*)d_in[5];
  float*               out   = (float*)d_out;

  // workspace layout (~1.3 MB)
  float4* xpack = (float4*)d_ws;            // B*N float4
  float4* ypack = xpack + (size_t)B_ * N_;  // B*N float4
  float*  f     = (float*)(ypack + (size_t)B_ * N_);
  float*  u     = f;                        // B*N (u_eff)
  float*  v     = f + 32768;                // B*N (v_eff)
  float*  na    = f + 65536;                // B
  float*  nb    = f + 65552;                // B
  float*  acc   = f + 65568;                // 1

  k_prep  <<<128, 256, 0, stream>>>(x, y, xpack, ypack);
  k_counts<<<B_,  32,  0, stream>>>(xm, ym, na, nb);
  k_zero  <<<128, 256, 0, stream>>>(u, v, ym, acc);

  for (int it = 0; it < NITER; ++it) {
    // u-step: rows = x, cols = y (uses v_eff)
    k_step<<<256, 256, 0, stream>>>(xpack, xm, na, ypack, v, u);
    // v-step: rows = y, cols = x (uses freshly written u_eff)
    k_step<<<256, 256, 0, stream>>>(ypack, ym, nb, xpack, u, v);
  }

  k_pc <<<256, 256, 0, stream>>>(xpack, ypack, u, v, acc);
  k_fin<<<1, 32, 0, stream>>>(kls, epoch, acc, out);
}